// attentive_feature_fusion_44985487458841
// MI455X (gfx1250) — compile-verified
//
#include <hip/hip_runtime.h>

typedef float v2f __attribute__((ext_vector_type(2)));
typedef float v4f __attribute__((ext_vector_type(4)));
typedef float v8f __attribute__((ext_vector_type(8)));

#define CDIM   512          // C
#define XDIM   1024         // 2C
#define ROWS   16           // rows per tile (one wave per tile)
#define PITCH  516          // LDS row pitch in floats: 512 + 4 pad -> conflict-free WMMA A feed

// One wave (32 lanes) per 16-row tile.
//  1) async-copy g-half (16x512 f32) global->LDS, coalesced B128 (ASYNCcnt)
//  2) 128x V_WMMA_F32_16X16X4_F32 with A fed from LDS (bank-conflict-free)
//  3) repeat for v-half into the same 33KB buffer
//  4) 2-way softmax scalars via shuffles
//  5) scale+store, re-reading g/v coalesced (L2-resident)
__global__ __launch_bounds__(32) void attentive_fusion_kernel(
    const float* __restrict__ g, const float* __restrict__ v,
    const float* __restrict__ W, const float* __restrict__ bias,
    float* __restrict__ out, int nrows)
{
    __shared__ float half_tile[ROWS * PITCH];   // 33,024 bytes

    const int lane = threadIdx.x & 31;
    const int row0 = blockIdx.x * ROWS;
    if (row0 >= nrows) return;

    const int m  = lane & 15;   // A row / B column / D column index
    const int hk = lane >> 4;   // K sub-pair selector: K = 4kk + 2*hk + {0,1}

    // B-matrix: only columns N=0,1 carry W rows 0/1; all other columns zero.
    const float  bmask = (m < 2) ? 1.0f : 0.0f;
    const float* wrow  = W + (size_t)((m < 2) ? m : 0) * XDIM;

    v8f acc = {};

    auto do_half = [&](const float* __restrict__ src, const float* __restrict__ wbase) {
        // ---- stage 16 rows x 512 floats, fully coalesced async B128 ----
#pragma unroll
        for (int r = 0; r < ROWS; ++r) {
            const float*   sp   = src + (size_t)(row0 + r) * CDIM + lane * 4;
            const unsigned ldsb = (unsigned)(size_t)&half_tile[r * PITCH + lane * 4];
#pragma unroll
            for (int i = 0; i < 4; ++i) {
                // INST_OFFSET applies to both global and LDS addresses; both
                // advance 512B per step within a row.
                asm volatile("global_load_async_to_lds_b128 %0, %1, off offset:%2"
                             :: "v"(ldsb),
                                "v"((unsigned long long)(size_t)sp),
                                "i"(i * 512)
                             : "memory");
            }
        }
        asm volatile("s_wait_asynccnt 0x0" ::: "memory");

        // ---- 128 WMMAs: A from LDS (ds_load_b64, conflict-free), B from W (L0-hit) ----
        const float* ap = &half_tile[m * PITCH + 2 * hk];
        const float* wp = wbase + 2 * hk;
#pragma unroll 8
        for (int kk = 0; kk < CDIM / 4; ++kk) {
            v2f a  = *(const v2f*)(ap + 4 * kk);
            v2f w  = *(const v2f*)(wp + 4 * kk);
            v2f bm = w * bmask;
            acc = __builtin_amdgcn_wmma_f32_16x16x4_f32(
                false, a, false, bm, (short)0, acc, false, false);
        }
        // WAR guard: all LDS reads retired before the next async overwrite.
        asm volatile("s_wait_dscnt 0x0" ::: "memory");
    };

    do_half(g, wrow);           // x_cat cols    0..511  <-> W cols    0..511
    do_half(v, wrow + CDIM);    // x_cat cols 512..1023  <-> W cols 512..1023

    // ---- 2-way softmax per row: att0 = sigmoid((l0+b0)-(l1+b1)) ----
    const float b0 = bias[0];
    const float b1 = bias[1];
    float att0v[ROWS];
#pragma unroll
    for (int j = 0; j < 8; ++j) {
        float dj  = acc[j];
        float l0a = __shfl(dj, 0, 32);   // row j,   logit0
        float l1a = __shfl(dj, 1, 32);   // row j,   logit1
        float l0b = __shfl(dj, 16, 32);  // row j+8, logit0
        float l1b = __shfl(dj, 17, 32);  // row j+8, logit1
        att0v[j]     = 1.0f / (1.0f + __expf((l1a + b1) - (l0a + b0)));
        att0v[j + 8] = 1.0f / (1.0f + __expf((l1b + b1) - (l0b + b0)));
    }

    // ---- Phase 2: coalesced scale + store (g/v rows are L2-resident) ----
#pragma unroll
    for (int r = 0; r < ROWS; ++r) {
        const float a0 = att0v[r];
        const float a1 = 1.0f - a0;
        const float* gp = g + (size_t)(row0 + r) * CDIM;
        const float* vp = v + (size_t)(row0 + r) * CDIM;
        float*       op = out + (size_t)(row0 + r) * XDIM;
#pragma unroll
        for (int i = 0; i < 4; ++i) {
            const int col = i * 128 + lane * 4;    // 16B-aligned, fully coalesced
            v4f xg = *(const v4f*)(gp + col);
            v4f xv = *(const v4f*)(vp + col);
            *(v4f*)(op + col)        = xg * a0;
            *(v4f*)(op + CDIM + col) = xv * a1;
        }
    }
}

extern "C" void kernel_launch(void* const* d_in, const int* in_sizes, int n_in,
                              void* d_out, int out_size, void* d_ws, size_t ws_size,
                              hipStream_t stream) {
    const float* g    = (const float*)d_in[0];   // [B, C]
    const float* v    = (const float*)d_in[1];   // [B, C]
    const float* W    = (const float*)d_in[2];   // [2, 2C]
    const float* bias = (const float*)d_in[3];   // [2]
    float* out        = (float*)d_out;           // [B, 2C]

    const int nrows  = in_sizes[0] / CDIM;              // B = 65536
    const int blocks = (nrows + ROWS - 1) / ROWS;       // 4096 tiles, 1 wave each

    attentive_fusion_kernel<<<blocks, 32, 0, stream>>>(g, v, W, bias, out, nrows);
}